// Quantize_3_12756052869874
// MI455X (gfx1250) — compile-verified
//
#include <hip/hip_runtime.h>

#define NROWS 65536        // B*T
#define NEMB  1024
#define NDIM  64
#define QELEMS (NROWS * NDIM)   // 4194304
#define NBLOCKS (NROWS / 8)     // 8192 blocks, 8 rows (one wave32 each) per block

#if defined(__has_builtin)
#  if __has_builtin(__builtin_amdgcn_global_load_async_to_lds_b128)
#    define HAVE_ASYNC_LDS 1
#  endif
#  if __has_builtin(__builtin_amdgcn_s_wait_asynccnt)
#    define HAVE_WAIT_ASYNC 1
#  endif
#endif

typedef int v4i __attribute__((vector_size(16)));
typedef __attribute__((address_space(1))) v4i* gptr_v4i;
typedef __attribute__((address_space(3))) v4i* lptr_v4i;

// ---------------------------------------------------------------------------
// Fused kernel: per-row argmax over 1024 floats -> codebook gather ->
// quantize write + MSE partial.  One wave32 per row, 8 rows per block.
// The 32KB tile of `ind` is staged into LDS with CDNA5 async global->LDS DMA
// (ASYNCcnt), then reduced with wave32 shuffle butterflies.
// ---------------------------------------------------------------------------
__global__ __launch_bounds__(256) void vq_fused(const float* __restrict__ ind,
                                                const float* __restrict__ input,
                                                const float* __restrict__ embedT,
                                                float* __restrict__ out_q,
                                                float* __restrict__ out_ind,
                                                double* __restrict__ partials) {
    __shared__ __align__(16) float lds[8 * NEMB];   // 32 KB staging for ind
    __shared__ double wsum[8];
    const int tid = threadIdx.x;
    const float* gsrc = ind + (size_t)blockIdx.x * (8 * NEMB);

#if defined(HAVE_ASYNC_LDS)
    #pragma unroll
    for (int q = 0; q < 8; ++q) {
        unsigned off = (unsigned)tid * 16u + (unsigned)q * 4096u;   // bytes
        __builtin_amdgcn_global_load_async_to_lds_b128(
            (gptr_v4i)((const char*)gsrc + off),
            (lptr_v4i)((char*)lds + off),
            0, 0);
    }
#  if defined(HAVE_WAIT_ASYNC)
    __builtin_amdgcn_s_wait_asynccnt(0);
#  else
    asm volatile("s_wait_asynccnt 0" ::: "memory");
#  endif
#else
    #pragma unroll
    for (int q = 0; q < 8; ++q) {
        unsigned off = (unsigned)tid * 4u + (unsigned)q * 1024u;    // floats
        *(float4*)&lds[off] = *(const float4*)&gsrc[off];
    }
#endif
    __syncthreads();

    const int wave = tid >> 5;
    const int lane = tid & 31;
    const int r    = blockIdx.x * 8 + wave;       // global row id
    const float* row = &lds[wave * NEMB];

    // ---- per-lane scan (32 elements, increasing index order) ----
    float best = -3.402823466e38f;
    int bidx = 0;
    #pragma unroll
    for (int j = 0; j < 8; ++j) {
        int pos = lane * 4 + j * 128;
        float4 v = *(const float4*)&row[pos];
        if (v.x > best) { best = v.x; bidx = pos;     }
        if (v.y > best) { best = v.y; bidx = pos + 1; }
        if (v.z > best) { best = v.z; bidx = pos + 2; }
        if (v.w > best) { best = v.w; bidx = pos + 3; }
    }
    // ---- wave32 butterfly: max on (val, then smaller index) — associative;
    //      all lanes converge to the row argmax ----
    #pragma unroll
    for (int off = 16; off > 0; off >>= 1) {
        float ov = __shfl_xor(best, off, 32);
        int   oi = __shfl_xor(bidx, off, 32);
        if (ov > best || (ov == best && oi < bidx)) { best = ov; bidx = oi; }
    }
    if (lane == 0) out_ind[r] = (float)bidx;

    // ---- gather codebook row, write quantize, accumulate MSE partial ----
    // lane l covers d = 2l, 2l+1  (coalesced 256B per wave from embedT/input)
    float2 q = *(const float2*)&embedT[bidx * NDIM + lane * 2];
    float2 x = *(const float2*)&input[r * NDIM + lane * 2];
    *(float2*)&out_q[r * NDIM + lane * 2] = q;
    float dx = q.x - x.x, dy = q.y - x.y;
    double acc = (double)(dx * dx) + (double)(dy * dy);

    // ---- deterministic block reduction ----
    #pragma unroll
    for (int off = 16; off > 0; off >>= 1)
        acc += __shfl_xor(acc, off, 32);
    if (lane == 0) wsum[wave] = acc;
    __syncthreads();
    if (tid == 0) {
        double s = 0.0;
        #pragma unroll
        for (int w = 0; w < 8; ++w) s += wsum[w];
        partials[blockIdx.x] = s;
    }
}

// ---------------------------------------------------------------------------
// transpose embed [64][1024] -> embedT [1024][64] so gathers are coalesced
// ---------------------------------------------------------------------------
__global__ __launch_bounds__(256) void transpose_embed(const float* __restrict__ embed,
                                                       float* __restrict__ embedT) {
    int flat = blockIdx.x * 256 + threadIdx.x;   // 0..65535
    int d = flat & 63;
    int e = flat >> 6;
    embedT[flat] = embed[d * NEMB + e];
}

// ---------------------------------------------------------------------------
// deterministic final reduction of 8192 block partials -> diff
// ---------------------------------------------------------------------------
__global__ __launch_bounds__(256) void finalize_diff(const double* __restrict__ partials,
                                                     float* __restrict__ out_diff) {
    __shared__ double s[256];
    int t = threadIdx.x;
    double a = 0.0;
    #pragma unroll
    for (int k = 0; k < NBLOCKS / 256; ++k)      // 32 terms, fixed order
        a += partials[t + k * 256];
    s[t] = a;
    __syncthreads();
    for (int off = 128; off > 0; off >>= 1) {
        if (t < off) s[t] += s[t + off];
        __syncthreads();
    }
    if (t == 0) *out_diff = (float)(s[0] / (double)QELEMS);
}

// ---------------------------------------------------------------------------
extern "C" void kernel_launch(void* const* d_in, const int* in_sizes, int n_in,
                              void* d_out, int out_size, void* d_ws, size_t ws_size,
                              hipStream_t stream) {
    const float* input = (const float*)d_in[0];   // [32,2048,64]
    const float* ind   = (const float*)d_in[1];   // [65536,1024]
    const float* embed = (const float*)d_in[2];   // [64,1024]
    (void)in_sizes; (void)n_in; (void)out_size; (void)ws_size;

    float* out_q    = (float*)d_out;              // [4194304]
    float* out_diff = out_q + QELEMS;             // [1]
    float* out_ind  = out_q + QELEMS + 1;         // [65536] as float

    char*   ws       = (char*)d_ws;
    float*  embedT   = (float*)ws;                // 256 KB
    double* partials = (double*)(ws + 262144);    // 64 KB (8192 doubles)

    transpose_embed<<<256, 256, 0, stream>>>(embed, embedT);
    vq_fused<<<NBLOCKS, 256, 0, stream>>>(ind, input, embedT, out_q, out_ind, partials);
    finalize_diff<<<1, 256, 0, stream>>>(partials, out_diff);
}